// Model_11888469476028
// MI455X (gfx1250) — compile-verified
//
#include <hip/hip_runtime.h>

// Problem constants (from reference)
#define B_TOT   4096
#define C_INCH  64
#define L_IN    128
#define C_OUT   16
#define L_UP    256
#define L_PATCH 252
#define KFOLD   5

typedef __attribute__((ext_vector_type(2))) float v2f;
typedef __attribute__((ext_vector_type(8))) float v8f;

#define WX 132   // padded LDS row stride for 16x128 parity weights (conflict-free b64 A loads)
#define XS 132   // padded LDS row stride for x (multiple of 4 dwords -> b128-aligned async copy)
#define FS 17    // padded stride for 16x16 scratch / mix weights

__launch_bounds__(256)
__global__ void fused_deconv_fold_mix(const float* __restrict__ x,
                                      const float* __restrict__ deconv_w,
                                      const float* __restrict__ deconv_b,
                                      const float* __restrict__ patch_w,
                                      const float* __restrict__ mix_w,
                                      const float* __restrict__ mix_b,
                                      float* __restrict__ out) {
  // lx layout: 4-dword zero guard, then 64 rows of [128 data | 4 zero pad].
  // Data element (c,s) at lxd[c*XS + s]; s=-1 resolves to the previous row's
  // zero pad (or the guard for c=0), s=128 to this row's pad -> branch-free edges.
  __shared__ float lx[4 + C_INCH * XS];    // 33,808 B
  __shared__ float lw[2][C_OUT * WX];      // 16,896 B: parity weights (even: taps 1,3 ; odd: taps 0,2)
  __shared__ float lmix[C_OUT * FS];       //  1,088 B: mix_w[o'][o]
  __shared__ float lpw[C_OUT * KFOLD];
  __shared__ float ldb[C_OUT];
  __shared__ float lmb[C_OUT];
  __shared__ float lfold[8][C_OUT * FS];   //  8,704 B: per-wave folded 16x16 scratch

  const int tid = threadIdx.x;
  const int b   = blockIdx.x;
  float* lxd = lx + 4;

  // ---- zero the guard + per-row pads (ds stores; async loads never touch these) ----
  if (tid < 4) lx[tid] = 0.f;
  if (tid >= 4 && tid < 4 + C_INCH) {
    int c = tid - 4;
    float4 z = {0.f, 0.f, 0.f, 0.f};
    *(float4*)(lxd + c * XS + L_IN) = z;
  }

  // ---- async-copy x[b] (8192 floats) to LDS: 8 x global_load_async_to_lds_b128/thread ----
  const float* xb = x + (size_t)b * (C_INCH * L_IN);
  #pragma unroll
  for (int i = 0; i < 8; ++i) {
    int v4 = tid + i * 256;                // vec4 index, 0..2047 (coalesced)
    int c = v4 >> 5, j = (v4 & 31) * 4;    // 32 vec4 per 128-elem row
    unsigned ldsa = (unsigned)(size_t)(lxd + c * XS + j);
    const float* ga = xb + c * L_IN + j;
    asm volatile("global_load_async_to_lds_b128 %0, %1, off"
                 :: "v"(ldsa), "v"(ga) : "memory");
  }

  // ---- build parity weight matrices: W_p[o][c]    = w[o,c, p?0:1]
  //                                    W_p[o][64+c] = w[o,c, p?2:3] ----
  #pragma unroll
  for (int i = 0; i < 8; ++i) {
    int idx = tid + i * 256;               // 0..2047
    int o = idx >> 7, cc = idx & 127;
    int c = cc & 63, hi64 = cc >> 6;
    lw[0][o * WX + cc] = deconv_w[(o * 64 + c) * 4 + (hi64 ? 3 : 1)];
    lw[1][o * WX + cc] = deconv_w[(o * 64 + c) * 4 + (hi64 ? 2 : 0)];
  }
  lmix[(tid >> 4) * FS + (tid & 15)] = mix_w[tid & 255];   // 256 entries
  if (tid < C_OUT * KFOLD) lpw[tid] = patch_w[tid];
  if (tid < C_OUT) { ldb[tid] = deconv_b[tid]; lmb[tid] = mix_b[tid]; }

  asm volatile("s_wait_asynccnt 0x0" ::: "memory");   // async tile resident in LDS
  __syncthreads();

  // ---- per-wave WMMA work: wave owns s' in [16*wave, 16*wave+16) ----
  const int wave = tid >> 5;
  const int lane = tid & 31;
  const int n    = lane & 15;    // N column (position j) / A row m
  const int hi   = lane >> 4;    // lane half selects K sub-pair and M high half
  const int s0   = wave * 16;
  float* fold = lfold[wave];

  v8f res[2];

  #pragma unroll
  for (int p = 0; p < 2; ++p) {            // p=0: even t=2s', p=1: odd t=2s'+1
    const float* Arow = lw[p] + n * WX + 2 * hi;   // a = {Arow[4kk], Arow[4kk+1]}
    v8f acc = {0.f, 0.f, 0.f, 0.f, 0.f, 0.f, 0.f, 0.f};

    // deconv GEMM: (16x128)x(128x16); split at the k=64 parity flip so the
    // B operand advances by a pure pointer induction (one v_add per step).
    const float* bp = lxd + (2 * hi) * XS + (s0 + n + p);        // k<64: s = s0+n+p
    #pragma unroll 8
    for (int kk = 0; kk < 16; ++kk) {
      v2f a;  a.x  = Arow[4 * kk]; a.y  = Arow[4 * kk + 1];
      v2f bf; bf.x = bp[0];        bf.y = bp[XS];
      bp += 4 * XS;
      acc = __builtin_amdgcn_wmma_f32_16x16x4_f32(
          false, a, false, bf, (short)0, acc, false, false);
    }
    bp = lxd + (2 * hi) * XS + (s0 + n + p - 1);                 // k>=64: s = s0+n+p-1
    #pragma unroll 8
    for (int kk = 16; kk < 32; ++kk) {
      v2f a;  a.x  = Arow[4 * kk]; a.y  = Arow[4 * kk + 1];
      v2f bf; bf.x = bp[0];        bf.y = bp[XS];
      bp += 4 * XS;
      acc = __builtin_amdgcn_wmma_f32_16x16x4_f32(
          false, a, false, bf, (short)0, acc, false, false);
    }

    // epilogue: bias + hardsigmoid + fold mask -> per-wave LDS scratch (B layout source)
    #pragma unroll
    for (int r = 0; r < 8; ++r) {
      int o = r + 8 * hi;
      int t = 2 * (s0 + n) + p;
      float v = acc[r] + ldb[o];
      v = fminf(fmaxf(v * (1.f / 6.f) + 0.5f, 0.f), 1.f);   // hardsigmoid
      float m = 0.f;
      #pragma unroll
      for (int k = 0; k < KFOLD; ++k) {
        int u = t - k;
        m += (u >= 0 && u < L_PATCH) ? lpw[o * KFOLD + k] : 0.f;
      }
      fold[o * FS + n] = v * m;
    }
    __builtin_amdgcn_wave_barrier();        // same-wave DS ops are in-order; pin scheduling

    // mix GEMM: (16x16)x(16x16), K=16 -> 4 WMMA steps
    v8f m2 = {0.f, 0.f, 0.f, 0.f, 0.f, 0.f, 0.f, 0.f};
    #pragma unroll
    for (int kk = 0; kk < 4; ++kk) {
      int kb = kk * 4 + 2 * hi;
      v2f a;
      a.x = lmix[n * FS + kb];
      a.y = lmix[n * FS + kb + 1];
      v2f bf;
      bf.x = fold[kb * FS + n];
      bf.y = fold[(kb + 1) * FS + n];
      m2 = __builtin_amdgcn_wmma_f32_16x16x4_f32(
          false, a, false, bf, (short)0, m2, false, false);
    }
    __builtin_amdgcn_wave_barrier();        // fold scratch reused by next phase

    #pragma unroll
    for (int r = 0; r < 8; ++r) m2[r] += lmb[r + 8 * hi];
    res[p] = m2;
  }

  // ---- store: interleave even/odd into contiguous b64 pairs: out[b][o][2(s0+n) .. +1] ----
  float* ob = out + (size_t)b * (C_OUT * L_UP);
  #pragma unroll
  for (int r = 0; r < 8; ++r) {
    int o = r + 8 * hi;
    int t = 2 * (s0 + n);
    v2f pr;
    pr.x = res[0][r];
    pr.y = res[1][r];
    *(v2f*)(ob + o * L_UP + t) = pr;
  }
}

extern "C" void kernel_launch(void* const* d_in, const int* in_sizes, int n_in,
                              void* d_out, int out_size, void* d_ws, size_t ws_size,
                              hipStream_t stream) {
  (void)in_sizes; (void)n_in; (void)d_ws; (void)ws_size; (void)out_size;
  const float* x  = (const float*)d_in[0];
  const float* dw = (const float*)d_in[1];
  const float* db = (const float*)d_in[2];
  const float* pw = (const float*)d_in[3];
  const float* mw = (const float*)d_in[4];
  const float* mb = (const float*)d_in[5];
  fused_deconv_fold_mix<<<B_TOT, 256, 0, stream>>>(x, dw, db, pw, mw, mb, (float*)d_out);
}